// CharacteristicFunctionNetwork_13975823581943
// MI455X (gfx1250) — compile-verified
//
#include <hip/hip_runtime.h>
#include <math.h>

typedef __attribute__((ext_vector_type(2))) float v2f;
typedef __attribute__((ext_vector_type(8))) float v8f;

#define ORDER 3
#define POINTS 16
#define NNODES 2048
#define D1 64
#define D2 32
#define POOL1 32
#define POOL2 8
#define LABELS 10
#define KIN (ORDER * POINTS)   // 48

// ---------------------------------------------------------------------------
// Kernel 1: feats[o,p,i] = mean_j cos(wm[o,p] * adj[o,i,j]); write transposed
// into ms[i, o*16+p].  One block per (o,i) row; trans-op bound.
// ---------------------------------------------------------------------------
__global__ __launch_bounds__(256) void cf_feats_kernel(
    const float* __restrict__ adj,   // [3,2048,2048]
    const float* __restrict__ wm,    // [3,16]
    float* __restrict__ ms)          // [2048,48]
{
    const int i   = blockIdx.x;      // node row
    const int o   = blockIdx.y;      // order
    const int tid = threadIdx.x;

    __shared__ float wsh[POINTS];
    if (tid < POINTS) wsh[tid] = wm[o * POINTS + tid];
    __syncthreads();

    float wl[POINTS];
#pragma unroll
    for (int p = 0; p < POINTS; ++p) wl[p] = wsh[p];

    const float* __restrict__ row = adj + ((size_t)o * NNODES + i) * NNODES;

    float acc[POINTS];
#pragma unroll
    for (int p = 0; p < POINTS; ++p) acc[p] = 0.0f;

    for (int j = tid; j < NNODES; j += 256) {
        __builtin_prefetch(&row[j + 256], 0, 0);   // global_prefetch_b8
        float a = row[j];
#pragma unroll
        for (int p = 0; p < POINTS; ++p) acc[p] += __cosf(wl[p] * a);
    }

    // wave32 shuffle reduce, then cross-wave via LDS
    __shared__ float red[POINTS * 8];
    const int lane = tid & 31;
    const int wv   = tid >> 5;
#pragma unroll
    for (int p = 0; p < POINTS; ++p) {
        float v = acc[p];
#pragma unroll
        for (int off = 16; off > 0; off >>= 1) v += __shfl_down(v, off, 32);
        if (lane == 0) red[p * 8 + wv] = v;
    }
    __syncthreads();
    if (tid < POINTS) {
        float s = 0.0f;
#pragma unroll
        for (int w = 0; w < 8; ++w) s += red[tid * 8 + w];
        ms[i * KIN + o * POINTS + tid] = s * (1.0f / (float)NNODES);
    }
}

// ---------------------------------------------------------------------------
// WMMA f32 16x16x4 fragment helpers (wave32).
// A 16x4: lanes 0-15 hold {K0,K1} of row=lane; lanes 16-31 hold {K2,K3}.
// B 4x16: v0 = {row K0 (lanes 0-15), row K1 (lanes 16-31)}; v1 = {K2, K3}.
// C/D 16x16: VGPR r: lanes 0-15 -> M=r, N=lane; lanes 16-31 -> M=r+8.
// ---------------------------------------------------------------------------
__device__ __forceinline__ v2f load_a_f32(const float* p, int ld, int kbase) {
    int lane = threadIdx.x & 31;
    int r  = lane & 15;
    int kk = kbase + ((lane >> 4) << 1);
    v2f a;
    a[0] = p[r * ld + kk];
    a[1] = p[r * ld + kk + 1];
    return a;
}

__device__ __forceinline__ v2f load_b_f32(const float* p, int ld, int kbase,
                                          int cbase, int maxc) {
    int lane = threadIdx.x & 31;
    int col  = cbase + (lane & 15);
    int k0   = kbase + (lane >> 4);
    v2f b;
    bool ok = (col < maxc);
    b[0] = ok ? p[k0 * ld + col]       : 0.0f;
    b[1] = ok ? p[(k0 + 2) * ld + col] : 0.0f;
    return b;
}

__device__ __forceinline__ v8f wmma_f32(v2f a, v2f b, v8f c) {
    return __builtin_amdgcn_wmma_f32_16x16x4_f32(
        /*neg_a=*/false, a, /*neg_b=*/false, b,
        /*c_mod=*/(short)0, c, /*reuse_a=*/false, /*reuse_b=*/false);
}

// ---------------------------------------------------------------------------
// Kernel 2: per 16-row tile (one wave): h1=relu(ms@w1+b1), h2=relu(h1@w2+b2),
// abs=tanh(h2@p1+pb1), s=abs@p2+pb2.  Writes h2 and s to workspace.
// ---------------------------------------------------------------------------
__global__ __launch_bounds__(32) void mlp_wmma_kernel(
    const float* __restrict__ ms,   // [2048,48]
    const float* __restrict__ w1, const float* __restrict__ b1,   // [48,64],[64]
    const float* __restrict__ w2, const float* __restrict__ b2,   // [64,32],[32]
    const float* __restrict__ p1, const float* __restrict__ pb1,  // [32,32],[32]
    const float* __restrict__ p2, const float* __restrict__ pb2,  // [32,8],[8]
    float* __restrict__ h2out,      // [2048,32]
    float* __restrict__ sout)       // [2048,8]
{
    const int row0  = blockIdx.x * 16;
    const int lane  = threadIdx.x & 31;
    const int col15 = lane & 15;
    const int mbase = (lane >> 4) << 3;   // 0 or 8

    __shared__ float stA[16 * D1];        // h1 tile / abstract tile
    __shared__ float stB[16 * D2];        // h2 tile

    // ---- h1 = relu(ms_tile @ w1 + b1) : [16,48]@[48,64] ----
    const float* msrow = ms + (size_t)row0 * KIN;
#pragma unroll
    for (int nt = 0; nt < D1 / 16; ++nt) {
        v8f acc = {};
#pragma unroll
        for (int kc = 0; kc < KIN / 4; ++kc) {
            v2f a = load_a_f32(msrow, KIN, kc * 4);
            v2f b = load_b_f32(w1, D1, kc * 4, nt * 16, D1);
            acc = wmma_f32(a, b, acc);
        }
        int col = nt * 16 + col15;
        float bias = b1[col];
#pragma unroll
        for (int r = 0; r < 8; ++r)
            stA[(mbase + r) * D1 + col] = fmaxf(acc[r] + bias, 0.0f);
    }
    __syncthreads();

    // ---- h2 = relu(h1 @ w2 + b2) : [16,64]@[64,32] ----
#pragma unroll
    for (int nt = 0; nt < D2 / 16; ++nt) {
        v8f acc = {};
#pragma unroll
        for (int kc = 0; kc < D1 / 4; ++kc) {
            v2f a = load_a_f32(stA, D1, kc * 4);
            v2f b = load_b_f32(w2, D2, kc * 4, nt * 16, D2);
            acc = wmma_f32(a, b, acc);
        }
        int col = nt * 16 + col15;
        float bias = b2[col];
#pragma unroll
        for (int r = 0; r < 8; ++r) {
            float v = fmaxf(acc[r] + bias, 0.0f);
            stB[(mbase + r) * D2 + col] = v;
            h2out[(size_t)(row0 + mbase + r) * D2 + col] = v;
        }
    }
    __syncthreads();

    // ---- abstract = tanh(h2 @ p1 + pb1) : [16,32]@[32,32] ----
#pragma unroll
    for (int nt = 0; nt < POOL1 / 16; ++nt) {
        v8f acc = {};
#pragma unroll
        for (int kc = 0; kc < D2 / 4; ++kc) {
            v2f a = load_a_f32(stB, D2, kc * 4);
            v2f b = load_b_f32(p1, POOL1, kc * 4, nt * 16, POOL1);
            acc = wmma_f32(a, b, acc);
        }
        int col = nt * 16 + col15;
        float bias = pb1[col];
#pragma unroll
        for (int r = 0; r < 8; ++r)
            stA[(mbase + r) * POOL1 + col] = tanhf(acc[r] + bias);
    }
    __syncthreads();

    // ---- s = abstract @ p2 + pb2 : [16,32]@[32,8] (B zero-padded to 16) ----
    {
        v8f acc = {};
#pragma unroll
        for (int kc = 0; kc < POOL1 / 4; ++kc) {
            v2f a = load_a_f32(stA, POOL1, kc * 4);
            v2f b = load_b_f32(p2, POOL2, kc * 4, 0, POOL2);
            acc = wmma_f32(a, b, acc);
        }
        if (col15 < POOL2) {
            float bias = pb2[col15];
#pragma unroll
            for (int r = 0; r < 8; ++r)
                sout[(size_t)(row0 + mbase + r) * POOL2 + col15] = acc[r] + bias;
        }
    }
}

// ---------------------------------------------------------------------------
// Kernel 3: att = softmax(s, axis=0); g = att.T @ h2; logits = g@cw+cb;
// out = log_softmax(logits).  Single workgroup.
// ---------------------------------------------------------------------------
__global__ __launch_bounds__(256) void pool_head_kernel(
    const float* __restrict__ h2,   // [2048,32]
    const float* __restrict__ s,    // [2048,8]
    const float* __restrict__ cw,   // [256,10]
    const float* __restrict__ cb,   // [10]
    float* __restrict__ out)        // [10]
{
    const int tid = threadIdx.x;
    __shared__ float red[256];
    __shared__ float cmax[POOL2];
    __shared__ float cden[POOL2];
    __shared__ float g[POOL2 * D2];
    __shared__ float logits[LABELS];

    for (int c = 0; c < POOL2; ++c) {
        float m = -INFINITY;
        for (int i = tid; i < NNODES; i += 256) m = fmaxf(m, s[i * POOL2 + c]);
        red[tid] = m; __syncthreads();
        for (int o = 128; o > 0; o >>= 1) {
            if (tid < o) red[tid] = fmaxf(red[tid], red[tid + o]);
            __syncthreads();
        }
        if (tid == 0) cmax[c] = red[0];
        __syncthreads();
        float mm = cmax[c];
        float ssum = 0.0f;
        for (int i = tid; i < NNODES; i += 256) ssum += __expf(s[i * POOL2 + c] - mm);
        red[tid] = ssum; __syncthreads();
        for (int o = 128; o > 0; o >>= 1) {
            if (tid < o) red[tid] += red[tid + o];
            __syncthreads();
        }
        if (tid == 0) cden[c] = red[0];
        __syncthreads();
    }

    // g[c,d] = sum_i softmax(s)[i,c] * h2[i,d]; one (c,d) per thread
    {
        int c = tid >> 5;            // 0..7
        int d = tid & 31;            // 0..31
        float mm = cmax[c], inv = 1.0f / cden[c];
        float acc = 0.0f;
        for (int i = 0; i < NNODES; ++i)
            acc += __expf(s[i * POOL2 + c] - mm) * inv * h2[i * D2 + d];
        g[tid] = acc;
    }
    __syncthreads();

    if (tid < LABELS) {
        float acc = cb[tid];
        for (int k = 0; k < POOL2 * D2; ++k) acc += g[k] * cw[k * LABELS + tid];
        logits[tid] = acc;
    }
    __syncthreads();

    if (tid == 0) {
        float m = logits[0];
        for (int l = 1; l < LABELS; ++l) m = fmaxf(m, logits[l]);
        float ssum = 0.0f;
        for (int l = 0; l < LABELS; ++l) ssum += expf(logits[l] - m);
        float lse = m + logf(ssum);
        for (int l = 0; l < LABELS; ++l) out[l] = logits[l] - lse;
    }
}

// ---------------------------------------------------------------------------
extern "C" void kernel_launch(void* const* d_in, const int* in_sizes, int n_in,
                              void* d_out, int out_size, void* d_ws, size_t ws_size,
                              hipStream_t stream) {
    const float* adj = (const float*)d_in[0];   // [3,2048,2048]
    const float* wm  = (const float*)d_in[1];   // [3,16]
    const float* w1  = (const float*)d_in[2];   // [48,64]
    const float* b1  = (const float*)d_in[3];   // [64]
    const float* w2  = (const float*)d_in[4];   // [64,32]
    const float* b2  = (const float*)d_in[5];   // [32]
    const float* p1  = (const float*)d_in[6];   // [32,32]
    const float* pb1 = (const float*)d_in[7];   // [32]
    const float* p2  = (const float*)d_in[8];   // [32,8]
    const float* pb2 = (const float*)d_in[9];   // [8]
    const float* cw  = (const float*)d_in[10];  // [256,10]
    const float* cb  = (const float*)d_in[11];  // [10]
    float* outp = (float*)d_out;

    float* ms  = (float*)d_ws;                  // [2048,48]
    float* h2  = ms + (size_t)NNODES * KIN;     // [2048,32]
    float* sc  = h2 + (size_t)NNODES * D2;      // [2048,8]

    cf_feats_kernel<<<dim3(NNODES, ORDER), 256, 0, stream>>>(adj, wm, ms);
    mlp_wmma_kernel<<<NNODES / 16, 32, 0, stream>>>(ms, w1, b1, w2, b2,
                                                    p1, pb1, p2, pb2, h2, sc);
    pool_head_kernel<<<1, 256, 0, stream>>>(h2, sc, cw, cb, outp);
}